// Decoder_58514634440787
// MI455X (gfx1250) — compile-verified
//
#include <hip/hip_runtime.h>
#include <hip/hip_bf16.h>
#include <stdint.h>

#define IMG_SIZE 196608        // 3*256*256
#define N_BATCH  8
#define NKNOT    33
#define NSEG     32
#define THREADS  256
#define PIX_PER_BLOCK (THREADS * 4)   // 1024 pixels per block, float4 per thread

// CDNA5 async global->LDS path (ASYNCcnt), with safe fallback if the
// toolchain doesn't expose the builtins.
#if __has_builtin(__builtin_amdgcn_global_load_async_to_lds_b32) && \
    __has_builtin(__builtin_amdgcn_s_wait_asynccnt)
#define HAVE_ASYNC_LDS 1
#else
#define HAVE_ASYNC_LDS 0
#endif

#if HAVE_ASYNC_LDS
typedef __attribute__((address_space(1))) int* gint_ptr;
typedef __attribute__((address_space(3))) int* lint_ptr;
#endif

__global__ __launch_bounds__(THREADS)
void decoder_piecewise_kernel(const float* __restrict__ seg_x,
                              const float* __restrict__ seg_y,
                              float* __restrict__ out) {
    __shared__ float sxs[NKNOT];   // raw x knots (staged)
    __shared__ float sys[NKNOT];   // raw y knots (staged)
    __shared__ float lX[NKNOT];    // fixed (running-max) x knots
    __shared__ float lY[NKNOT];    // y carried with running max
    __shared__ float lR[NSEG];     // per-segment slope

    const int b   = blockIdx.y;
    const int tid = threadIdx.x;

    // ---- Stage this batch's knot rows into LDS (async load path, ASYNCcnt) ----
    if (tid < NKNOT) {
#if HAVE_ASYNC_LDS
        __builtin_amdgcn_global_load_async_to_lds_b32(
            (gint_ptr)(void*)(seg_x + b * NKNOT + tid),
            (lint_ptr)(void*)(&sxs[tid]), 0, 0);
        __builtin_amdgcn_global_load_async_to_lds_b32(
            (gint_ptr)(void*)(seg_y + b * NKNOT + tid),
            (lint_ptr)(void*)(&sys[tid]), 0, 0);
#else
        sxs[tid] = seg_x[b * NKNOT + tid];
        sys[tid] = seg_y[b * NKNOT + tid];
#endif
    }
#if HAVE_ASYNC_LDS
    __builtin_amdgcn_s_wait_asynccnt(0);   // drain this wave's ASYNCcnt
#endif
    __syncthreads();

    // ---- Fix x-order: running max of x with y carried (ties take later) ----
    // Thread t computes the prefix up to knot t (broadcast LDS reads).
    if (tid < NKNOT) {
        float x = sxs[0], y = sys[0];
        for (int i = 1; i <= tid; ++i) {
            float xi = sxs[i], yi = sys[i];
            if (xi >= x) { x = xi; y = yi; }
        }
        lX[tid] = x;
        lY[tid] = y;
    }
    __syncthreads();

    // ---- Per-segment slope with divider==0 -> 1e-4 rule ----
    if (tid < NSEG) {
        float d = lX[tid + 1] - lX[tid];
        if (d == 0.0f) d = 0.0001f;
        lR[tid] = (lY[tid + 1] - lY[tid]) / d;
    }
    __syncthreads();

    // ---- Each thread: 4 consecutive pixels, one coalesced B128 store ----
    const int   s0   = (blockIdx.x * THREADS + tid) * 4;
    const float xin0 = (float)(s0 + 1) / (float)IMG_SIZE;

    // Branchless binary search: idx = #{k in [0,32] : lX[k] <= xin0}.
    // Then n = clamp(idx-1, 0, 31) is the unique live segment (handles
    // below-first / above-last extrapolation and zero-width duplicates).
    int idx = 0;
#pragma unroll
    for (int step = 32; step > 0; step >>= 1) {
        int j = idx + step;
        if (j <= NKNOT && lX[j - 1] <= xin0) idx = j;
    }
    int n = idx - 1;
    n = (n < 0) ? 0 : ((n > NSEG - 1) ? (NSEG - 1) : n);

    float r[4];
#pragma unroll
    for (int i = 0; i < 4; ++i) {
        float xin = (float)(s0 + i + 1) / (float)IMG_SIZE;
        // x_in is monotone within the quad: advance segment as needed
        // (loop handles runs of duplicate knots correctly).
        while (n < NSEG - 1 && xin >= lX[n + 1]) ++n;
        r[i] = fmaf(lR[n], xin - lX[n], lY[n]);
    }

    float4 o;
    o.x = r[0]; o.y = r[1]; o.z = r[2]; o.w = r[3];
    *(float4*)(out + (size_t)b * IMG_SIZE + s0) = o;   // global_store_b128
}

extern "C" void kernel_launch(void* const* d_in, const int* in_sizes, int n_in,
                              void* d_out, int out_size, void* d_ws, size_t ws_size,
                              hipStream_t stream) {
    (void)in_sizes; (void)n_in; (void)out_size; (void)d_ws; (void)ws_size;
    const float* seg_x = (const float*)d_in[0];  // [8, 33] f32
    const float* seg_y = (const float*)d_in[1];  // [8, 33] f32
    float*       out   = (float*)d_out;          // [8, 196608] f32

    dim3 grid(IMG_SIZE / PIX_PER_BLOCK, N_BATCH, 1);  // (192, 8)
    decoder_piecewise_kernel<<<grid, THREADS, 0, stream>>>(seg_x, seg_y, out);
}